// GATBranch_34419867910757
// MI455X (gfx1250) — compile-verified
//
#include <hip/hip_runtime.h>

#define NEG_SLOPE 0.2f

typedef __attribute__((ext_vector_type(2))) float v2f;
typedef __attribute__((ext_vector_type(8))) float v8f;

// ---- float atomic max via sign-split integer atomics (works with -inf init) ----
__device__ __forceinline__ void atomic_max_float(float* addr, float val) {
    if (val >= 0.0f) {
        atomicMax((int*)addr, __float_as_int(val));
    } else {
        atomicMin((unsigned int*)addr, __float_as_uint(val));
    }
}

// =====================  C[M x 64] = A[M x K] @ B[K x 64]  =====================
// B (<=128x64 = 32KB) is staged into LDS once per block with CDNA5 async
// global->LDS copies (ASYNCcnt / s_wait_asynccnt), then each of the 8 waves
// computes a 16x16 output tile with V_WMMA_F32_16X16X4_F32, K stepped by 4.
// A 16x4 layout: lanes 0-15 hold K=0(.x),1(.y); lanes 16-31 hold K=2(.x),3(.y).
// B 4x16 layout mirrors it. C/D: VGPR v -> row m0+v / m0+v+8, col n0+(lane&15).
__global__ void gemm_wmma_f32(const float* __restrict__ A, const float* __restrict__ B,
                              float* __restrict__ C, int M, int K) {
    const int NOUT = 64;
    __shared__ float bs[128 * 64];               // 32 KB of the WGP's 320 KB LDS

    // ---- async stage B[K x 64] into LDS (b128 = 4 floats per op) ----
    {
        const unsigned lds_base = (unsigned)(unsigned long long)(void*)bs;
        const int nvec = (K * NOUT) >> 2;        // # of float4 chunks
        for (int t = threadIdx.x; t < nvec; t += (int)blockDim.x) {
            unsigned laddr = lds_base + (unsigned)t * 16u;
            unsigned long long gaddr = (unsigned long long)(const void*)(B + (long long)t * 4);
            asm volatile("global_load_async_to_lds_b128 %0, %1, off"
                         :: "v"(laddr), "v"(gaddr) : "memory");
        }
        asm volatile("s_wait_asynccnt 0x0" ::: "memory");
    }
    __syncthreads();

    const int wave = threadIdx.x >> 5;           // 8 waves / block
    const int lane = threadIdx.x & 31;
    const int mtile = blockIdx.x * 2 + (wave >> 2); // 2 m-tiles per block
    const int ntile = wave & 3;                     // 64/16 = 4 n-tiles
    if (mtile * 16 < M) {                           // wave-uniform: EXEC stays full
        const int m0 = mtile * 16;
        const int n0 = ntile * 16;
        const int lrow  = lane & 15;
        const int khalf = (lane >> 4) << 1;         // 0 or 2

        const float* arow = A + (long long)(m0 + lrow) * K;
        const float* bcol = bs + n0 + lrow;         // LDS-resident B

        v8f c = {};
        for (int k0 = 0; k0 < K; k0 += 4) {
            v2f a, b;
            a.x = arow[k0 + khalf + 0];
            a.y = arow[k0 + khalf + 1];
            b.x = bcol[(k0 + khalf + 0) * NOUT];
            b.y = bcol[(k0 + khalf + 1) * NOUT];
            c = __builtin_amdgcn_wmma_f32_16x16x4_f32(false, a, false, b,
                                                      (short)0, c, false, false);
        }

        const int rbase = m0 + ((lane >> 4) << 3);
        float* crow = C + (long long)rbase * NOUT + n0 + lrow;
#pragma unroll
        for (int v = 0; v < 8; v++) {
            crow[(long long)v * NOUT] = c[v];
        }
    }
}

// ================= per-node attention scores: s = h . a ======================
__global__ void node_scores(const float* __restrict__ h,
                            const float* __restrict__ a_src,
                            const float* __restrict__ a_dst,
                            float* __restrict__ ssrc, float* __restrict__ sdst, int N) {
    int i = blockIdx.x * blockDim.x + threadIdx.x;
    if (i >= N) return;
    const float* row = h + (long long)i * 64;
    float s0 = 0.f, s1 = 0.f;
#pragma unroll
    for (int k = 0; k < 64; k++) {
        float v = row[k];
        s0 += v * a_src[k];
        s1 += v * a_dst[k];
    }
    ssrc[i] = s0;
    sdst[i] = s1;
}

// ============== init: out accumulator = 0, segmax = -inf, segsum = 0 =========
__global__ void init_seg(float* __restrict__ outacc, float* __restrict__ mseg,
                         float* __restrict__ dseg, int N) {
    long long gid = (long long)blockIdx.x * blockDim.x + threadIdx.x;
    if (gid < (long long)N * 64) outacc[gid] = 0.0f;
    if (gid < N) {
        mseg[gid] = -__builtin_inff();
        dseg[gid] = 0.0f;
    }
}

// ======= pass 1: logits + leaky_relu, store, segment-max over dst ============
__global__ void edge_logit_max(const int* __restrict__ src, const int* __restrict__ dst,
                               const float* __restrict__ ssrc, const float* __restrict__ sdst,
                               float* __restrict__ lr_out, float* __restrict__ mseg,
                               int E, int Etot) {
    int e = blockIdx.x * blockDim.x + threadIdx.x;
    if (e >= Etot) return;
    int s, d;
    if (e < E) { s = src[e]; d = dst[e]; } else { s = e - E; d = s; }
    float t  = ssrc[s] + sdst[d];
    float lr = t > 0.0f ? t : NEG_SLOPE * t;
    lr_out[e] = lr;
    atomic_max_float(&mseg[d], lr);
}

// ======= pass 2: ex = exp(e - m[dst]); segment-sum over dst ==================
__global__ void edge_expsum(const int* __restrict__ dst, float* __restrict__ ex,
                            const float* __restrict__ mseg, float* __restrict__ dseg,
                            int E, int Etot) {
    int e = blockIdx.x * blockDim.x + threadIdx.x;
    if (e >= Etot) return;
    int d = (e < E) ? dst[e] : (e - E);
    float v = expf(ex[e] - mseg[d]);
    ex[e] = v;
    atomicAdd(&dseg[d], v);
}

// ======= pass 3: out[dst] += alpha * h[src]; one wave per edge (64 feats) ====
__global__ void edge_aggregate(const int* __restrict__ src, const int* __restrict__ dst,
                               const float* __restrict__ h, const float* __restrict__ ex,
                               const float* __restrict__ dseg, float* __restrict__ outacc,
                               int E, int Etot) {
    long long gid = (long long)blockIdx.x * blockDim.x + threadIdx.x;
    int e    = (int)(gid >> 5);
    int lane = (int)(gid & 31);
    if (e >= Etot) return;
    int s, d;
    if (e < E) { s = src[e]; d = dst[e]; } else { s = e - E; d = s; }
    float alpha = ex[e] / (dseg[d] + 1e-16f);
    const float2 hv = *(const float2*)(h + (long long)s * 64 + 2 * lane); // one b64 gather
    float* od = outacc + (long long)d * 64 + 2 * lane;
    atomicAdd(&od[0], alpha * hv.x);
    atomicAdd(&od[1], alpha * hv.y);
}

// ============== finalize: hpost = relu(outacc + bias) ========================
__global__ void finalize_relu_bias(const float* __restrict__ acc, const float* __restrict__ b,
                                   float* __restrict__ hpost, long long total) {
    long long gid = (long long)blockIdx.x * blockDim.x + threadIdx.x;
    if (gid >= total) return;
    int f = (int)(gid & 63);
    float v = acc[gid] + b[f];
    hpost[gid] = v > 0.0f ? v : 0.0f;
}

// ============== deterministic two-stage mean over nodes ======================
__global__ void mean_pool_partial(const float* __restrict__ h, float* __restrict__ partial, int N) {
    __shared__ float sm[256];
    int f = threadIdx.x & 63;
    int g = threadIdx.x >> 6;              // 4 stripes per block
    int stripe = blockIdx.x * 4 + g;       // 256 stripes total (64 blocks)
    float s = 0.0f;
    for (int i = stripe; i < N; i += 256) s += h[(long long)i * 64 + f];
    sm[threadIdx.x] = s;
    __syncthreads();
    if (g == 0) partial[blockIdx.x * 64 + f] = sm[f] + sm[64 + f] + sm[128 + f] + sm[192 + f];
}

__global__ void mean_pool_final(const float* __restrict__ partial, float* __restrict__ out, int N) {
    __shared__ float sm[256];
    int f = threadIdx.x & 63;
    int g = threadIdx.x >> 6;
    float s = 0.0f;
    for (int p = g; p < 64; p += 4) s += partial[p * 64 + f];
    sm[threadIdx.x] = s;
    __syncthreads();
    if (g == 0) out[f] = (sm[f] + sm[64 + f] + sm[128 + f] + sm[192 + f]) / (float)N;
}

extern "C" void kernel_launch(void* const* d_in, const int* in_sizes, int n_in,
                              void* d_out, int out_size, void* d_ws, size_t ws_size,
                              hipStream_t stream) {
    const float* x    = (const float*)d_in[0];
    const int*   ei   = (const int*)d_in[1];      // [2, E] int32
    const float* W1   = (const float*)d_in[2];
    const float* a1s  = (const float*)d_in[3];
    const float* a1d  = (const float*)d_in[4];
    const float* b1   = (const float*)d_in[5];
    const float* W2   = (const float*)d_in[6];
    const float* a2s  = (const float*)d_in[7];
    const float* a2d  = (const float*)d_in[8];
    const float* b2   = (const float*)d_in[9];

    const int N    = in_sizes[0] / 128;   // 50000
    const int E    = in_sizes[1] / 2;     // 800000
    const int Etot = E + N;               // + self loops
    const int* src = ei;
    const int* dst = ei + E;

    // workspace layout (floats)
    float* ws      = (float*)d_ws;
    float* h       = ws;                        // N*64  (WMMA gemm output)
    float* hpost   = h       + (size_t)N * 64;  // N*64  (post bias+relu, next layer input)
    float* outacc  = hpost   + (size_t)N * 64;  // N*64  (scatter accumulator)
    float* ssrc    = outacc  + (size_t)N * 64;  // N
    float* sdst    = ssrc    + N;               // N
    float* mseg    = sdst    + N;               // N
    float* dseg    = mseg    + N;               // N
    float* ex      = dseg    + N;               // Etot
    float* partial = ex      + Etot;            // 64*64

    const long long totNF = (long long)N * 64;
    const int mtiles      = (N + 15) / 16;
    const int gemm_blocks = (mtiles + 1) / 2;           // 2 m-tiles per block
    const int nblk        = (N + 255) / 256;
    const int nfblk       = (int)((totNF + 255) / 256);
    const int eblk        = (Etot + 255) / 256;
    const int aggblk      = (int)(((long long)Etot * 32 + 255) / 256);

    // ----------------- layer 1 -----------------
    gemm_wmma_f32<<<gemm_blocks, 256, 0, stream>>>(x, W1, h, N, 128);
    node_scores<<<nblk, 256, 0, stream>>>(h, a1s, a1d, ssrc, sdst, N);
    init_seg<<<nfblk, 256, 0, stream>>>(outacc, mseg, dseg, N);
    edge_logit_max<<<eblk, 256, 0, stream>>>(src, dst, ssrc, sdst, ex, mseg, E, Etot);
    edge_expsum<<<eblk, 256, 0, stream>>>(dst, ex, mseg, dseg, E, Etot);
    edge_aggregate<<<aggblk, 256, 0, stream>>>(src, dst, h, ex, dseg, outacc, E, Etot);
    finalize_relu_bias<<<nfblk, 256, 0, stream>>>(outacc, b1, hpost, totNF);

    // ----------------- layer 2 -----------------
    gemm_wmma_f32<<<gemm_blocks, 256, 0, stream>>>(hpost, W2, h, N, 64);
    node_scores<<<nblk, 256, 0, stream>>>(h, a2s, a2d, ssrc, sdst, N);
    init_seg<<<nfblk, 256, 0, stream>>>(outacc, mseg, dseg, N);
    edge_logit_max<<<eblk, 256, 0, stream>>>(src, dst, ssrc, sdst, ex, mseg, E, Etot);
    edge_expsum<<<eblk, 256, 0, stream>>>(dst, ex, mseg, dseg, E, Etot);
    edge_aggregate<<<aggblk, 256, 0, stream>>>(src, dst, h, ex, dseg, outacc, E, Etot);
    finalize_relu_bias<<<nfblk, 256, 0, stream>>>(outacc, b2, hpost, totNF);

    // ----------------- global mean pool (two-stage, deterministic) -----------
    mean_pool_partial<<<64, 256, 0, stream>>>(hpost, partial, N);
    mean_pool_final<<<1, 256, 0, stream>>>(partial, (float*)d_out, N);
}